// GATConv_29386166239304
// MI455X (gfx1250) — compile-verified
//
#include <hip/hip_runtime.h>
#include <math.h>

// ---------------------------------------------------------------------------
// GATConv forward for MI455X (gfx1250, wave32).
//   feat [N,128] @ W [128,64] -> ft  (V_WMMA_F32_16X16X4_F32, fp32-exact)
//   el/er score reductions, edge segment-softmax via float atomics,
//   weighted scatter-add of messages (all L2-resident: ft = 12.8MB << 192MB L2)
// ---------------------------------------------------------------------------

typedef __attribute__((ext_vector_type(2))) float v2f;
typedef __attribute__((ext_vector_type(8))) float v8f;

#define IN_FEATS 128
#define HEADS    4
#define DHEAD    16
#define HD       64          // HEADS * DHEAD
#define NEG_SLOPE 0.2f

// float atomic max via int-bit ordering trick (valid with -inf init, mixed signs)
__device__ __forceinline__ void atomicMaxFloat(float* addr, float val) {
  if (val >= 0.0f) {
    atomicMax((int*)addr, __float_as_int(val));
  } else {
    atomicMin((unsigned int*)addr, __float_as_uint(val));
  }
}

// ---------------------------------------------------------------------------
// Phase 0: out = bias (broadcast), m = -inf, s = 0
// ---------------------------------------------------------------------------
__global__ __launch_bounds__(256) void gat_init(float* __restrict__ out,
                                                const float* __restrict__ bias,
                                                float* __restrict__ m,
                                                float* __restrict__ s,
                                                int nNodes) {
  int i = blockIdx.x * blockDim.x + threadIdx.x;
  int totalOut = nNodes * HD;
  if (i < totalOut) out[i] = bias[i & (HD - 1)];
  if (i < nNodes * HEADS) {
    m[i] = -INFINITY;
    s[i] = 0.0f;
  }
}

// ---------------------------------------------------------------------------
// Phase 1: ft = feat @ W  via V_WMMA_F32_16X16X4_F32.
// One wave -> 16 rows x 64 cols (4 column tiles == 4 heads).
// ISA f32 fragment layout (wave32):
//   A 16x4 : lane<16  -> row=lane,    v0=K0, v1=K1
//            lane>=16 -> row=lane-16, v0=K2, v1=K3
//   B 4x16 : lane<16  -> col=lane,    v0=K0, v1=K1
//            lane>=16 -> col=lane-16, v0=K2, v1=K3
//   C 16x16: vgpr r, lane<16 -> (M=r, N=lane); lane>=16 -> (M=r+8, N=lane-16)
// ---------------------------------------------------------------------------
__global__ __launch_bounds__(256) void gat_proj_wmma(
    const float* __restrict__ feat, const float* __restrict__ W,
    float* __restrict__ ft, int nNodes, int nTiles) {
  const int lane = threadIdx.x & 31;
  const int tile = blockIdx.x * (blockDim.x >> 5) + (threadIdx.x >> 5);
  if (tile >= nTiles) return;           // wave-uniform: EXEC stays all-1s

  const int row0 = tile * 16;
  const int mrow = lane & 15;
  const int kq   = lane >> 4;           // 0 -> K pair {0,1}, 1 -> K pair {2,3}

  int arow = row0 + mrow;
  if (arow >= nNodes) arow = nNodes - 1;   // clamp (no EXEC divergence)
  const float2* __restrict__ apf2 =
      (const float2*)(feat + (size_t)arow * IN_FEATS) + kq;

  v8f acc0 = {}, acc1 = {}, acc2 = {}, acc3 = {};

#pragma unroll 4
  for (int k = 0; k < IN_FEATS; k += 4) {
    float2 av = apf2[k >> 1];           // feat[arow, k+2*kq .. k+2*kq+1]
    v2f a; a[0] = av.x; a[1] = av.y;

    const float* wr0 = W + (size_t)(k + 2 * kq) * HD;   // W row (k+2kq)
    const float* wr1 = wr0 + HD;                        // W row (k+2kq+1)
    v2f b0; b0[0] = wr0[mrow];      b0[1] = wr1[mrow];
    v2f b1; b1[0] = wr0[mrow + 16]; b1[1] = wr1[mrow + 16];
    v2f b2; b2[0] = wr0[mrow + 32]; b2[1] = wr1[mrow + 32];
    v2f b3; b3[0] = wr0[mrow + 48]; b3[1] = wr1[mrow + 48];

    acc0 = __builtin_amdgcn_wmma_f32_16x16x4_f32(false, a, false, b0,
                                                 (short)0, acc0, false, false);
    acc1 = __builtin_amdgcn_wmma_f32_16x16x4_f32(false, a, false, b1,
                                                 (short)0, acc1, false, false);
    acc2 = __builtin_amdgcn_wmma_f32_16x16x4_f32(false, a, false, b2,
                                                 (short)0, acc2, false, false);
    acc3 = __builtin_amdgcn_wmma_f32_16x16x4_f32(false, a, false, b3,
                                                 (short)0, acc3, false, false);
  }

#pragma unroll
  for (int r = 0; r < 8; ++r) {
    int row = row0 + r + 8 * kq;
    if (row < nNodes) {
      float* o = ft + (size_t)row * HD + mrow;
      o[0]  = acc0[r];
      o[16] = acc1[r];
      o[32] = acc2[r];
      o[48] = acc3[r];
    }
  }
}

// ---------------------------------------------------------------------------
// Phase 2: el[n,h] = <ft[n,h,:], attn_l[h,:]>, er likewise.
// ---------------------------------------------------------------------------
__global__ __launch_bounds__(256) void gat_scores(
    const float* __restrict__ ft, const float* __restrict__ attn_l,
    const float* __restrict__ attn_r, float* __restrict__ el,
    float* __restrict__ er, int NH) {
  int i = blockIdx.x * blockDim.x + threadIdx.x;
  if (i >= NH) return;
  int h = i & (HEADS - 1);
  const float* f  = ft + (size_t)(i >> 2) * HD + h * DHEAD;
  const float* al = attn_l + h * DHEAD;
  const float* ar = attn_r + h * DHEAD;
  float sl = 0.0f, sr = 0.0f;
#pragma unroll
  for (int d = 0; d < DHEAD; ++d) {
    float v = f[d];
    sl += v * al[d];
    sr += v * ar[d];
  }
  el[i] = sl;
  er[i] = sr;
}

// ---------------------------------------------------------------------------
// Phase 3: per-(edge,head) logit -> segment max into m[dst,h]
// ---------------------------------------------------------------------------
__global__ __launch_bounds__(256) void gat_edge_max(
    const int* __restrict__ src, const int* __restrict__ dst,
    const float* __restrict__ el, const float* __restrict__ er,
    float* __restrict__ m, int EH) {
  int i = blockIdx.x * blockDim.x + threadIdx.x;
  if (i >= EH) return;
  int e = i >> 2, h = i & (HEADS - 1);
  int didx = dst[e] * HEADS + h;
  float x = el[src[e] * HEADS + h] + er[didx];
  x = (x > 0.0f) ? x : NEG_SLOPE * x;       // leaky_relu
  atomicMaxFloat(&m[didx], x);
}

// ---------------------------------------------------------------------------
// Phase 4: ex = exp(logit - m[dst]); s[dst] += ex
// ---------------------------------------------------------------------------
__global__ __launch_bounds__(256) void gat_edge_expsum(
    const int* __restrict__ src, const int* __restrict__ dst,
    const float* __restrict__ el, const float* __restrict__ er,
    const float* __restrict__ m, float* __restrict__ s,
    float* __restrict__ ex, int EH) {
  int i = blockIdx.x * blockDim.x + threadIdx.x;
  if (i >= EH) return;
  int e = i >> 2, h = i & (HEADS - 1);
  int didx = dst[e] * HEADS + h;
  float x = el[src[e] * HEADS + h] + er[didx];
  x = (x > 0.0f) ? x : NEG_SLOPE * x;
  float v = expf(x - m[didx]);
  ex[i] = v;
  atomicAdd(&s[didx], v);
}

// ---------------------------------------------------------------------------
// Phase 5: out[dst, c] += ft[src, c] * (ex / s[dst]).  One thread per (e, c);
// 64 consecutive lanes handle one edge -> coalesced ft read + dense atomics.
// ---------------------------------------------------------------------------
__global__ __launch_bounds__(256) void gat_scatter(
    const int* __restrict__ src, const int* __restrict__ dst,
    const float* __restrict__ ft, const float* __restrict__ ex,
    const float* __restrict__ s, float* __restrict__ out, long long total) {
  long long i = (long long)blockIdx.x * blockDim.x + threadIdx.x;
  if (i >= total) return;
  int e = (int)(i >> 6);
  int c = (int)(i & (HD - 1));
  int h = c >> 4;
  int se = src[e], de = dst[e];
  float a = ex[e * HEADS + h] / s[de * HEADS + h];
  atomicAdd(&out[(size_t)de * HD + c], ft[(size_t)se * HD + c] * a);
}

// ---------------------------------------------------------------------------
extern "C" void kernel_launch(void* const* d_in, const int* in_sizes, int n_in,
                              void* d_out, int out_size, void* d_ws,
                              size_t ws_size, hipStream_t stream) {
  const float* feat   = (const float*)d_in[0];
  const int*   src    = (const int*)d_in[1];
  const int*   dst    = (const int*)d_in[2];
  const float* W      = (const float*)d_in[3];
  const float* attn_l = (const float*)d_in[4];
  const float* attn_r = (const float*)d_in[5];
  const float* bias   = (const float*)d_in[6];
  float* out = (float*)d_out;

  const int N = in_sizes[0] / IN_FEATS;
  const int E = in_sizes[1];

  // workspace layout (floats): ft | el | er | m | s | ex   (~29 MB total)
  float* ws = (float*)d_ws;
  float* ft = ws;
  float* el = ft + (size_t)N * HD;
  float* er = el + (size_t)N * HEADS;
  float* m  = er + (size_t)N * HEADS;
  float* s  = m  + (size_t)N * HEADS;
  float* ex = s  + (size_t)N * HEADS;

  // Phase 0: init out (bias), m (-inf), s (0)
  {
    int total = N * HD;
    gat_init<<<(total + 255) / 256, 256, 0, stream>>>(out, bias, m, s, N);
  }
  // Phase 1: WMMA projection
  {
    int nTiles = (N + 15) / 16;       // 3125 for N=50000 (exact)
    int blocks = (nTiles + 7) / 8;    // 8 waves / 256-thread block
    gat_proj_wmma<<<blocks, 256, 0, stream>>>(feat, W, ft, N, nTiles);
  }
  // Phase 2: attention scores
  {
    int NH = N * HEADS;
    gat_scores<<<(NH + 255) / 256, 256, 0, stream>>>(ft, attn_l, attn_r, el,
                                                     er, NH);
  }
  // Phase 3: segment max
  {
    int EH = E * HEADS;
    gat_edge_max<<<(EH + 255) / 256, 256, 0, stream>>>(src, dst, el, er, m, EH);
  }
  // Phase 4: exp + segment sum
  {
    int EH = E * HEADS;
    gat_edge_expsum<<<(EH + 255) / 256, 256, 0, stream>>>(src, dst, el, er, m,
                                                          s, ex, EH);
  }
  // Phase 5: weighted scatter-add
  {
    long long total = (long long)E * HD;
    int blocks = (int)((total + 255) / 256);
    gat_scatter<<<blocks, 256, 0, stream>>>(src, dst, ft, ex, s, out, total);
  }
}